// overAllRREA_37606733644139
// MI455X (gfx1250) — compile-verified
//
#include <hip/hip_runtime.h>
#include <hip/hip_bf16.h>
#include <math.h>

// Problem constants (match reference)
#define NN 100000
#define EE 1000000
#define RR 1000
#define DD 100
#define OUTC 600   // output row stride: concat(out_ent 300, out_rel 300)
#define EPW 16     // edges per wave in run-aware scatter kernels

typedef float v2f __attribute__((ext_vector_type(2)));
typedef float v8f __attribute__((ext_vector_type(8)));

// Monotone-in-both-domains atomic float max (seed with -inf).
static __device__ __forceinline__ void atomicMaxFloat(float* addr, float value) {
  if (value >= 0.0f)
    atomicMax((int*)addr, __float_as_int(value));
  else
    atomicMin((unsigned int*)addr, __float_as_uint(value));
}

// ---------------------------------------------------------------- fill
__global__ void fill_kernel(float* __restrict__ p, long long n, float v) {
  long long i = (long long)blockIdx.x * blockDim.x + threadIdx.x;
  if (i < n) p[i] = v;
}

// ---------------------------------------------------------------- rel normalize: rel_norm[r] = rel_emb[r]/max(||.||,1e-12)
__global__ void rel_norm_kernel(const float* __restrict__ rel_emb, float* __restrict__ rel_norm) {
  int r = __builtin_amdgcn_readfirstlane((int)((blockIdx.x * blockDim.x + threadIdx.x) >> 5));
  int lane = threadIdx.x & 31;
  if (r >= RR) return;
  const float* rr = rel_emb + (size_t)r * DD;
  float ss = 0.f;
  for (int k = lane; k < DD; k += 32) { float v = rr[k]; ss += v * v; }
#pragma unroll
  for (int off = 16; off > 0; off >>= 1) ss += __shfl_xor(ss, off, 32);
  float nrm = fmaxf(sqrtf(ss), 1e-12f);
  for (int k = lane; k < DD; k += 32) rel_norm[(size_t)r * DD + k] = rr[k] / nrm;
}

// ---------------------------------------------------------------- in-degree counts (shared by all segment means)
__global__ void count_kernel(const int* __restrict__ dst, float* __restrict__ cnt) {
  int e = blockIdx.x * blockDim.x + threadIdx.x;
  if (e < EE) atomicAdd(&cnt[dst[e]], 1.0f);
}

// ---------------------------------------------------------------- base feature scatter: accum[dst] += emb[rowidx]
// Run-aware: dst is sorted, so accumulate in registers while dst unchanged, flush once per run.
__global__ void scatter_base_kernel(const float* __restrict__ emb, const int* __restrict__ rowidx,
                                    const int* __restrict__ dst, float* __restrict__ accum) {
  int wave = __builtin_amdgcn_readfirstlane((int)((blockIdx.x * blockDim.x + threadIdx.x) >> 5));
  int lane = threadIdx.x & 31;
  int e0 = wave * EPW;
  if (e0 >= EE || lane >= 25) return;     // 25 lanes x float4 = 100 columns
  int eEnd = e0 + EPW; if (eEnd > EE) eEnd = EE;
  float ax = 0.f, ay = 0.f, az = 0.f, aw = 0.f;
  int cur = dst[e0];
  for (int e = e0; e < eEnd; ++e) {
    int row = rowidx[e];
    int d = dst[e];
    if (d != cur) {
      float* ar = accum + (size_t)cur * DD + lane * 4;
      atomicAdd(ar + 0, ax); atomicAdd(ar + 1, ay);
      atomicAdd(ar + 2, az); atomicAdd(ar + 3, aw);
      ax = ay = az = aw = 0.f;
      cur = d;
    }
    const float4 v = ((const float4*)(emb + (size_t)row * DD))[lane];
    ax += v.x; ay += v.y; az += v.z; aw += v.w;
  }
  float* ar = accum + (size_t)cur * DD + lane * 4;
  atomicAdd(ar + 0, ax); atomicAdd(ar + 1, ay);
  atomicAdd(ar + 2, az); atomicAdd(ar + 3, aw);
}

// ---------------------------------------------------------------- node: tanh(accum[/cnt]) -> feats & out column; reset state
// One thread per float4 (N*25 threads).
__global__ void node_kernel(float* __restrict__ accum, const float* __restrict__ cnt, int use_mean,
                            float* __restrict__ feats, float* __restrict__ outcol,
                            float* __restrict__ segmax, float* __restrict__ segsum) {
  long long i = (long long)blockIdx.x * blockDim.x + threadIdx.x;
  if (i >= (long long)NN * 25) return;
  int n = (int)(i / 25);
  int q = (int)(i - (long long)n * 25);
  float4 v = ((float4*)accum)[i];
  if (use_mean) {
    float inv = 1.0f / fmaxf(cnt[n], 1.0f);
    v.x *= inv; v.y *= inv; v.z *= inv; v.w *= inv;
  }
  float4 f;
  f.x = tanhf(v.x); f.y = tanhf(v.y); f.z = tanhf(v.z); f.w = tanhf(v.w);
  ((float4*)feats)[i] = f;
  *(float4*)(outcol + (size_t)n * OUTC + q * 4) = f;
  float4 z; z.x = 0.f; z.y = 0.f; z.z = 0.f; z.w = 0.f;
  ((float4*)accum)[i] = z;               // ready for next aggregation pass
  if (q == 0) { segmax[n] = -__builtin_inff(); segsum[n] = 0.0f; }
}

// ---------------------------------------------------------------- per-relation scalars via WMMA f32 16x16x4
// rdot[r*2+0] = rel_norm[r] . w_n ; rdot[r*2+1] = rel_norm[r] . w_r
// A tile: 16 relations x 4 K (lanes 0-15: K = k0+0,k0+1 ; lanes 16-31: K = k0+2,k0+3)
// B tile: 4 x 16, col 0 = w_n chunk, col 1 = w_r chunk, rest zero.
// B built with mask-FMAs so all loads are unconditional (no EXEC churn).
__global__ void rdot_wmma_kernel(const float* __restrict__ rel_norm,
                                 const float* __restrict__ attn_l,   // k[l]: w_s @0, w_n @DD, w_r @2DD
                                 float* __restrict__ rdot) {
  const int tile = blockIdx.x;                    // (RR+15)/16 tiles
  const int lane = threadIdx.x & 31;
  const int nCol = lane & 15;
  const int half = lane >> 4;                     // 0 or 1
  const int khalf = half * 2;                     // 0 or 2
  int mrow = tile * 16 + nCol;
  int mclamp = mrow < (RR - 1) ? mrow : (RR - 1); // keep loads in-bounds; stores are guarded
  const float* arow = rel_norm + (size_t)mclamp * DD;
  const float* w_n = attn_l + DD;
  const float* w_r = attn_l + 2 * DD;
  const float mn = (nCol == 0) ? 1.0f : 0.0f;     // lane masks, computed once
  const float mr = (nCol == 1) ? 1.0f : 0.0f;

  v8f c = {};
#pragma unroll 5
  for (int k0 = 0; k0 < DD; k0 += 4) {
    v2f a  = *(const v2f*)(arow + k0 + khalf);    // global_load_b64
    v2f wn = *(const v2f*)(w_n + k0 + khalf);     // global_load_b64 (8B aligned)
    v2f wr = *(const v2f*)(w_r + k0 + khalf);
    v2f b;
    b.x = wn.x * mn + wr.x * mr;                  // v_fma, no branches
    b.y = wn.y * mn + wr.y * mr;
    c = __builtin_amdgcn_wmma_f32_16x16x4_f32(false, a, false, b, (short)0, c, false, false);
  }

  // C layout: VGPR j, lanes 0-15 -> M=j ; lanes 16-31 -> M=8+j ; N = lane&15
  if (nCol < 2) {
#pragma unroll
    for (int j = 0; j < 8; ++j) {
      int m = tile * 16 + half * 8 + j;
      if (m < RR) rdot[(size_t)m * 2 + nCol] = c[j];
    }
  }
}

// ---------------------------------------------------------------- edge logits + segment max
// logit = fd.w_s + fs.w_n - 2*dot1*(rel.w_n) + (rel.w_r),  dot1 = fs.rel_norm
__global__ void edge_logits_kernel(const float* __restrict__ feats, const float* __restrict__ rel_norm,
                                   const float* __restrict__ attn_l, const float* __restrict__ rdot,
                                   const int* __restrict__ src, const int* __restrict__ dst,
                                   const int* __restrict__ erel,
                                   float* __restrict__ logit, float* __restrict__ dot1buf,
                                   float* __restrict__ segmax) {
  int e = __builtin_amdgcn_readfirstlane((int)((blockIdx.x * blockDim.x + threadIdx.x) >> 5));
  int lane = threadIdx.x & 31;
  if (e >= EE) return;
  int s = src[e], d = dst[e], r = erel[e];
  float p1 = 0.f, p2 = 0.f, p3 = 0.f;
  if (lane < 25) {
    const float4 a  = ((const float4*)(feats + (size_t)s * DD))[lane];
    const float4 b  = ((const float4*)(rel_norm + (size_t)r * DD))[lane];
    const float4 cd = ((const float4*)(feats + (size_t)d * DD))[lane];
    const float4 ws = ((const float4*)(attn_l))[lane];
    const float4 wn = ((const float4*)(attn_l + DD))[lane];
    p1 = a.x * b.x + a.y * b.y + a.z * b.z + a.w * b.w;
    p2 = a.x * wn.x + a.y * wn.y + a.z * wn.z + a.w * wn.w;
    p3 = cd.x * ws.x + cd.y * ws.y + cd.z * ws.z + cd.w * ws.w;
  }
#pragma unroll
  for (int off = 16; off > 0; off >>= 1) {
    p1 += __shfl_xor(p1, off, 32);
    p2 += __shfl_xor(p2, off, 32);
    p3 += __shfl_xor(p3, off, 32);
  }
  if (lane == 0) {
    float lg = p3 + p2 - 2.0f * p1 * rdot[(size_t)r * 2 + 0] + rdot[(size_t)r * 2 + 1];
    dot1buf[e] = p1;
    logit[e] = lg;
    atomicMaxFloat(&segmax[d], lg);
  }
}

// ---------------------------------------------------------------- exp + segment sum (logit buffer becomes exp buffer)
__global__ void edge_expsum_kernel(const int* __restrict__ dst, float* __restrict__ logit,
                                   const float* __restrict__ segmax, float* __restrict__ segsum) {
  int e = blockIdx.x * blockDim.x + threadIdx.x;
  if (e >= EE) return;
  int d = dst[e];
  float ex = expf(logit[e] - segmax[d]);
  logit[e] = ex;
  atomicAdd(&segsum[d], ex);
}

// ---------------------------------------------------------------- weighted reflected scatter:
// accum[dst] += att * (feats[src] - 2*dot1*rel_norm[erel])
// Run-aware over sorted dst, register accumulation, one atomic burst per run.
__global__ void edge_aggregate_kernel(const float* __restrict__ feats, const float* __restrict__ rel_norm,
                                      const int* __restrict__ src, const int* __restrict__ dst,
                                      const int* __restrict__ erel,
                                      const float* __restrict__ expv, const float* __restrict__ dot1buf,
                                      const float* __restrict__ segsum, float* __restrict__ accum) {
  int wave = __builtin_amdgcn_readfirstlane((int)((blockIdx.x * blockDim.x + threadIdx.x) >> 5));
  int lane = threadIdx.x & 31;
  int e0 = wave * EPW;
  if (e0 >= EE || lane >= 25) return;
  int eEnd = e0 + EPW; if (eEnd > EE) eEnd = EE;
  float ax = 0.f, ay = 0.f, az = 0.f, aw = 0.f;
  int cur = dst[e0];
  for (int e = e0; e < eEnd; ++e) {
    int s = src[e], d = dst[e], r = erel[e];
    if (d != cur) {
      float* ar = accum + (size_t)cur * DD + lane * 4;
      atomicAdd(ar + 0, ax); atomicAdd(ar + 1, ay);
      atomicAdd(ar + 2, az); atomicAdd(ar + 3, aw);
      ax = ay = az = aw = 0.f;
      cur = d;
    }
    float att = expv[e] / segsum[d];
    float two = 2.0f * dot1buf[e];
    const float4 a = ((const float4*)(feats + (size_t)s * DD))[lane];
    const float4 b = ((const float4*)(rel_norm + (size_t)r * DD))[lane];
    ax += att * (a.x - two * b.x);
    ay += att * (a.y - two * b.y);
    az += att * (a.z - two * b.z);
    aw += att * (a.w - two * b.w);
  }
  float* ar = accum + (size_t)cur * DD + lane * 4;
  atomicAdd(ar + 0, ax); atomicAdd(ar + 1, ay);
  atomicAdd(ar + 2, az); atomicAdd(ar + 3, aw);
}

// ================================================================ host
extern "C" void kernel_launch(void* const* d_in, const int* in_sizes, int n_in,
                              void* d_out, int out_size, void* d_ws, size_t ws_size,
                              hipStream_t stream) {
  const float* ent_emb = (const float*)d_in[0];   // N*D
  const float* rel_emb = (const float*)d_in[1];   // R*D
  const float* attn_e  = (const float*)d_in[2];   // DEPTH*3D
  const float* attn_r  = (const float*)d_in[3];   // DEPTH*3D
  const int*   src     = (const int*)d_in[4];     // E
  const int*   dst     = (const int*)d_in[5];     // E
  const int*   erel    = (const int*)d_in[6];     // E
  float* out = (float*)d_out;                     // N x 600
  float* ws  = (float*)d_ws;

  size_t o = 0;
  float* rel_n  = ws + o; o += (size_t)RR * DD;
  float* cnt    = ws + o; o += NN;
  float* segmax = ws + o; o += NN;
  float* segsum = ws + o; o += NN;
  float* rdot   = ws + o; o += (size_t)2 * RR;
  float* logit  = ws + o; o += EE;                // logits, then exp values
  float* dot1   = ws + o; o += EE;
  float* feats  = ws + o; o += (size_t)NN * DD;
  float* accum  = ws + o; o += (size_t)NN * DD;

  const int TB = 256;
  const long long ND = (long long)NN * DD;
  const int gridND4  = (int)(((long long)NN * 25 + TB - 1) / TB);       // float4 node kernel
  const int gridFill = (int)((ND + TB - 1) / TB);
  const int gridE    = (EE + TB - 1) / TB;
  const int gridEw   = (int)(((long long)EE * 32 + TB - 1) / TB);       // wave-per-edge
  const long long nScatWaves = (EE + EPW - 1) / EPW;
  const int gridScat = (int)((nScatWaves * 32 + TB - 1) / TB);          // wave-per-EPW-edges
  const int gridRw   = (RR * 32 + TB - 1) / TB;
  const int tilesR   = (RR + 15) / 16;

  // One-time setup
  rel_norm_kernel<<<gridRw, TB, 0, stream>>>(rel_emb, rel_n);
  fill_kernel<<<(NN + TB - 1) / TB, TB, 0, stream>>>(cnt, NN, 0.0f);
  fill_kernel<<<gridFill, TB, 0, stream>>>(accum, ND, 0.0f);
  fill_kernel<<<(NN + TB - 1) / TB, TB, 0, stream>>>(segmax, NN, -__builtin_inff());
  fill_kernel<<<(NN + TB - 1) / TB, TB, 0, stream>>>(segsum, NN, 0.0f);
  count_kernel<<<gridE, TB, 0, stream>>>(dst, cnt);

  for (int enc = 0; enc < 2; ++enc) {
    const float* emb    = enc ? rel_emb : ent_emb;
    const int*   rowidx = enc ? erel : src;
    const float* attn   = enc ? attn_r : attn_e;

    // base features: tanh(segment_mean(emb[rowidx], dst))
    scatter_base_kernel<<<gridScat, TB, 0, stream>>>(emb, rowidx, dst, accum);
    node_kernel<<<gridND4, TB, 0, stream>>>(accum, cnt, 1, feats,
                                            out + (size_t)enc * 300, segmax, segsum);

    for (int l = 0; l < 2; ++l) {
      const float* attn_l = attn + (size_t)l * 3 * DD;
      rdot_wmma_kernel<<<tilesR, 32, 0, stream>>>(rel_n, attn_l, rdot);
      edge_logits_kernel<<<gridEw, TB, 0, stream>>>(feats, rel_n, attn_l, rdot,
                                                    src, dst, erel, logit, dot1, segmax);
      edge_expsum_kernel<<<gridE, TB, 0, stream>>>(dst, logit, segmax, segsum);
      edge_aggregate_kernel<<<gridScat, TB, 0, stream>>>(feats, rel_n, src, dst, erel,
                                                         logit, dot1, segsum, accum);
      node_kernel<<<gridND4, TB, 0, stream>>>(accum, cnt, 0, feats,
                                              out + (size_t)enc * 300 + (size_t)(l + 1) * 100,
                                              segmax, segsum);
    }
  }
}